// mLSTMCell_70042326663309
// MI455X (gfx1250) — compile-verified
//
#include <hip/hip_runtime.h>
#include <math.h>

#define B_   64
#define DIM_ 1024

typedef float v2f __attribute__((ext_vector_type(2)));
typedef float v4f __attribute__((ext_vector_type(4)));
typedef float v8f __attribute__((ext_vector_type(8)));

// ---------------------------------------------------------------------------
// Kernel 1: gate GEMM  out[m,n] = (x @ W.T + b)[m,n] * scale
// x: [64,1024]  W: [1024,1024] row-major (out = x W^T => B-frag[k][n] = W[n][k])
// One wave computes one 16x16 output tile via V_WMMA_F32_16X16X4_F32,
// K-loop of 256 WMMAs.
// A-frag layout (16x4 f32): lane<16 holds K=0,1 ; lane>=16 holds K=2,3 (M=lane%16)
// B-frag layout (4x16 f32): lane<16 holds K=0,1 ; lane>=16 holds K=2,3 (N=lane%16)
// C/D layout (16x16 f32):  vgpr j -> M = j + 8*(lane>=16), N = lane%16
// ---------------------------------------------------------------------------
__global__ __launch_bounds__(32) void gate_gemm_wmma(
    const float* __restrict__ x, const float* __restrict__ W,
    const float* __restrict__ bias, float* __restrict__ out, float scale)
{
  const int n0   = blockIdx.x * 16;
  const int m0   = blockIdx.y * 16;
  const int lane = threadIdx.x & 31;
  const int lm   = lane & 15;
  const int hi   = lane >> 4;

  const float* arow = x + (m0 + lm) * DIM_ + 2 * hi;   // A[M=lm, K=2*hi + ...]
  const float* brow = W + (n0 + lm) * DIM_ + 2 * hi;   // B[K, N=lm] = W[n0+lm, K]

  v8f acc = {};
#pragma unroll 4
  for (int k = 0; k < DIM_; k += 4) {
    v2f a = *(const v2f*)(arow + k);
    v2f b = *(const v2f*)(brow + k);
    // (neg_a, A, neg_b, B, c_mod, C, reuse_a, reuse_b)
    acc = __builtin_amdgcn_wmma_f32_16x16x4_f32(false, a, false, b,
                                                (short)0, acc, false, false);
  }

  const float bn = bias[n0 + lm];
#pragma unroll
  for (int j = 0; j < 8; ++j) {
    const int M = j + 8 * hi;
    out[(m0 + M) * DIM_ + n0 + lm] = (acc[j] + bn) * scale;
  }
}

// ---------------------------------------------------------------------------
// Kernel 2: fused mLSTM state update, single streaming pass over C.
// One wave per row r of C[b] (32 lanes x 8 float4 = 1024 cols).
//   m_new = max(f_t + m, i_t);  i = exp(i_t - m_new);  f = exp(f_t + m - m_new)
//   C_new[b,r,c] = f_r * C[b,r,c] + (i_r * v_r) * k[b,c]     (NT load/store)
//   acc   = sum_c C_new[b,r,c] * q[b,c]   -> wave shuffle reduction
//   h[b,r] = sigmoid(o_t[b,r]) * acc
//   n_new[b,r] = f_r * n[b,r] + i_r * k[b,r];  m_new written per row
// ---------------------------------------------------------------------------
__global__ __launch_bounds__(256) void fused_update(
    const float* __restrict__ C,  const float* __restrict__ nvec,
    const float* __restrict__ mvec,
    const float* __restrict__ gq, const float* __restrict__ gk,
    const float* __restrict__ gv, const float* __restrict__ gi,
    const float* __restrict__ gf, const float* __restrict__ go,
    float* __restrict__ h, float* __restrict__ Cn,
    float* __restrict__ nn, float* __restrict__ mn)
{
  const int wave = threadIdx.x >> 5;
  const int lane = threadIdx.x & 31;
  const int b    = blockIdx.x >> 7;                 // 128 row-groups per batch
  const int r    = ((blockIdx.x & 127) << 3) + wave;
  const int br   = b * DIM_ + r;

  // per-row gate scalars (same loads across lanes -> single fetch, broadcast)
  const float it   = gi[br];
  const float ft   = gf[br];
  const float mm   = mvec[br];
  const float mnew = fmaxf(ft + mm, it);
  const float ig   = __expf(it - mnew);
  const float fg   = __expf(ft + mm - mnew);
  const float ivr  = ig * gv[br];

  const v4f* Crow  = (const v4f*)(C  + (size_t)br * DIM_);
  v4f*       Cnrow = (v4f*)      (Cn + (size_t)br * DIM_);
  const v4f* kv    = (const v4f*)(gk + (size_t)b * DIM_);
  const v4f* qv    = (const v4f*)(gq + (size_t)b * DIM_);

  float acc = 0.0f;
#pragma unroll
  for (int j = 0; j < 8; ++j) {
    const int c4 = j * 32 + lane;                   // contiguous 512B per instr
    v4f cold = __builtin_nontemporal_load(Crow + c4);
    v4f kk   = kv[c4];                              // cached: reused across rows
    v4f qq   = qv[c4];
    v4f cnew = fg * cold + ivr * kk;
    __builtin_nontemporal_store(cnew, Cnrow + c4);
    acc += cnew.x * qq.x + cnew.y * qq.y + cnew.z * qq.z + cnew.w * qq.w;
  }

  // wave32 reduction of the dot product
#pragma unroll
  for (int off = 16; off > 0; off >>= 1)
    acc += __shfl_xor(acc, off, 32);

  if (lane == 0) {
    const float og = go[br];
    const float sg = 1.0f / (1.0f + __expf(-og));
    h[br]  = sg * acc;
    nn[br] = fg * nvec[br] + ig * gk[br];
    mn[br] = mnew;
  }
}

// ---------------------------------------------------------------------------
extern "C" void kernel_launch(void* const* d_in, const int* in_sizes, int n_in,
                              void* d_out, int out_size, void* d_ws, size_t ws_size,
                              hipStream_t stream) {
  (void)in_sizes; (void)n_in; (void)out_size; (void)ws_size;

  const float* x = (const float*)d_in[0];
  const float* C = (const float*)d_in[1];
  const float* n = (const float*)d_in[2];
  const float* m = (const float*)d_in[3];

  const int G = B_ * DIM_;                 // 65536 elements per gate
  float* gates = (float*)d_ws;             // 6 * G floats = 1.5 MB scratch

  const float inv_sqrt_d = 0.03125f;       // 1/sqrt(1024)

  dim3 ggrid(DIM_ / 16, B_ / 16, 1);       // 64 x 4 tiles, one wave each
  for (int g = 0; g < 6; ++g) {            // order: q, k, v, i, f, o
    const float* W  = (const float*)d_in[4 + 2 * g];
    const float* bb = (const float*)d_in[5 + 2 * g];
    const float scale = (g == 1) ? inv_sqrt_d : 1.0f;
    gate_gemm_wmma<<<ggrid, 32, 0, stream>>>(x, W, bb, gates + (size_t)g * G, scale);
  }

  float* out = (float*)d_out;              // h | C_new | n_new | m_new
  float* h   = out;
  float* Cn  = out + G;
  float* nn  = Cn + (size_t)G * DIM_;
  float* mn  = nn + G;

  fused_update<<<dim3(B_ * (DIM_ / 8)), 256, 0, stream>>>(
      C, n, m,
      gates + 0 * (size_t)G, gates + 1 * (size_t)G, gates + 2 * (size_t)G,
      gates + 3 * (size_t)G, gates + 4 * (size_t)G, gates + 5 * (size_t)G,
      h, Cn, nn, mn);
}